// LevelWiseNodePooling_86672440033784
// MI455X (gfx1250) — compile-verified
//
#include <hip/hip_runtime.h>
#include <hip/hip_bf16.h>
#include <math.h>

typedef __attribute__((ext_vector_type(2))) float v2f;
typedef __attribute__((ext_vector_type(8))) float v8f;

#define NLVL 64
#define NDIM 256

__global__ __launch_bounds__(256)
void lvlpool_phase1(const float* __restrict__ emb,
                    const int*   __restrict__ dep,
                    float*       __restrict__ ws_sum,   // [G][64][256]
                    float*       __restrict__ ws_max,   // [G][64][256]
                    unsigned*    __restrict__ ws_cnt,   // [G][64]
                    int N, int chunk)
{
    __shared__ float    smax[NLVL * NDIM];
    __shared__ unsigned scnt[NLVL];

    const int tid = threadIdx.x;
    const int g   = blockIdx.x;

    for (int i = tid; i < NLVL * NDIM; i += 256) smax[i] = -INFINITY;
    if (tid < NLVL) scnt[tid] = 0u;
    __syncthreads();

    const int lane  = tid & 31;
    const int wave  = tid >> 5;
    const int half  = lane >> 4;        // 0: lanes 0-15 (K=0/1), 1: lanes 16-31 (K=2/3)
    const int fl    = lane & 15;
    const int fbase = wave * 32;        // this wave's 32-feature slice
    const int col0  = fbase + fl;       // feature column for ft=0 (ft=1 is col0+16)

    v8f c[4][2] = {};                   // [level tile][feature tile] 16x16 accumulators

    const int nbeg = g * chunk;
    const int nend = min(N, nbeg + chunk);

    // Per-lane induction variables (K-column assignment per WMMA A/B layout).
    int      i02 = nbeg + (half ? 2 : 0);
    int      i13 = nbeg + (half ? 3 : 1);
    unsigned o02 = (unsigned)i02 * NDIM + (unsigned)col0;
    unsigned o13 = (unsigned)i13 * NDIM + (unsigned)col0;

    int nb = nbeg;
    // ---- main loop: full batches of 4, no masking anywhere ----
    #pragma unroll 2
    for (; nb + 4 <= nend;
         nb += 4, i02 += 4, i13 += 4, o02 += 4 * NDIM, o13 += 4 * NDIM) {
        const unsigned d02 = (unsigned)dep[i02];
        const unsigned d13 = (unsigned)dep[i13];

        // B matrices (4x16 node x feature) loaded straight in WMMA layout.
        const float b00 = __builtin_nontemporal_load(emb + o02);        // ft0, K=0/2
        const float b01 = __builtin_nontemporal_load(emb + o02 + 16);   // ft1, K=0/2
        const float b10 = __builtin_nontemporal_load(emb + o13);        // ft0, K=1/3
        const float b11 = __builtin_nontemporal_load(emb + o13 + 16);   // ft1, K=1/3

        // Max pooling: native LDS float-max atomics (ds_max_num_f32),
        // shared address register + 64B immediate offsets.
        const unsigned m02 = d02 * NDIM + (unsigned)col0;
        const unsigned m13 = d13 * NDIM + (unsigned)col0;
        atomicMax(&smax[m02],      b00);
        atomicMax(&smax[m02 + 16], b01);
        atomicMax(&smax[m13],      b10);
        atomicMax(&smax[m13 + 16], b11);

        // Per-level counts: wave 0, lanes 0 and 16 cover nodes nb+0..3.
        if (wave == 0 && fl == 0) {
            atomicAdd(&scnt[d02], 1u);
            atomicAdd(&scnt[d13], 1u);
        }

        // Segment-sum as one-hot(depth)^T x emb via WMMA.
        #pragma unroll
        for (int lt = 0; lt < 4; ++lt) {
            const unsigned row = (unsigned)(lt * 16 + fl);
            v2f A;
            A.x = (d02 == row) ? 1.0f : 0.0f;
            A.y = (d13 == row) ? 1.0f : 0.0f;
            {
                v2f B; B.x = b00; B.y = b10;
                c[lt][0] = __builtin_amdgcn_wmma_f32_16x16x4_f32(
                    false, A, false, B, (short)0, c[lt][0], false, false);
            }
            {
                v2f B; B.x = b01; B.y = b11;
                c[lt][1] = __builtin_amdgcn_wmma_f32_16x16x4_f32(
                    false, A, false, B, (short)0, c[lt][1], false, false);
            }
        }
    }

    // ---- masked tail: at most one partial batch ----
    if (nb < nend) {
        const bool v02 = i02 < nend;
        const bool v13 = i13 < nend;
        const int c02 = v02 ? i02 : nbeg;
        const int c13 = v13 ? i13 : nbeg;

        const unsigned d02 = (unsigned)dep[c02];
        const unsigned d13 = (unsigned)dep[c13];

        float b0[2], b1[2];
        #pragma unroll
        for (int ft = 0; ft < 2; ++ft) {
            float t0 = emb[(size_t)c02 * NDIM + col0 + ft * 16];
            float t1 = emb[(size_t)c13 * NDIM + col0 + ft * 16];
            b0[ft] = v02 ? t0 : 0.0f;   // zero so one-hot * garbage can't make NaN
            b1[ft] = v13 ? t1 : 0.0f;
        }

        #pragma unroll
        for (int ft = 0; ft < 2; ++ft) {
            if (v02) atomicMax(&smax[d02 * NDIM + col0 + ft * 16], b0[ft]);
            if (v13) atomicMax(&smax[d13 * NDIM + col0 + ft * 16], b1[ft]);
        }
        if (wave == 0 && fl == 0) {
            if (v02) atomicAdd(&scnt[d02], 1u);
            if (v13) atomicAdd(&scnt[d13], 1u);
        }

        #pragma unroll
        for (int lt = 0; lt < 4; ++lt) {
            const unsigned row = (unsigned)(lt * 16 + fl);
            v2f A;
            A.x = (d02 == row) ? 1.0f : 0.0f;
            A.y = (d13 == row) ? 1.0f : 0.0f;
            #pragma unroll
            for (int ft = 0; ft < 2; ++ft) {
                v2f B; B.x = b0[ft]; B.y = b1[ft];
                c[lt][ft] = __builtin_amdgcn_wmma_f32_16x16x4_f32(
                    false, A, false, B, (short)0, c[lt][ft], false, false);
            }
        }
    }

    // Flush per-block sum tiles (C layout: VGPR j -> M=j (lanes 0-15), M=j+8 (16-31)).
    {
        float* outp = ws_sum + (size_t)g * (NLVL * NDIM);
        #pragma unroll
        for (int lt = 0; lt < 4; ++lt)
            #pragma unroll
            for (int ft = 0; ft < 2; ++ft)
                #pragma unroll
                for (int j = 0; j < 8; ++j) {
                    const int row = lt * 16 + j + half * 8;
                    const int col = col0 + ft * 16;
                    outp[row * NDIM + col] = c[lt][ft][j];
                }
    }

    __syncthreads();   // all LDS atomics visible

    for (int i = tid; i < NLVL * NDIM; i += 256)
        ws_max[(size_t)g * (NLVL * NDIM) + i] = smax[i];
    if (tid < NLVL)
        ws_cnt[g * NLVL + tid] = scnt[tid];
}

__global__ __launch_bounds__(256)
void lvlpool_phase2(const float*    __restrict__ ws_sum,
                    const float*    __restrict__ ws_max,
                    const unsigned* __restrict__ ws_cnt,
                    float* __restrict__ out, int G)
{
    const int t = blockIdx.x * 256 + threadIdx.x;   // 0 .. 16383
    const int l = t >> 8;
    const int f = t & 255;

    float s = 0.0f, m = -INFINITY;
    unsigned cnt = 0;
    for (int g = 0; g < G; ++g) {
        s += ws_sum[(size_t)g * (NLVL * NDIM) + t];
        m  = fmaxf(m, ws_max[(size_t)g * (NLVL * NDIM) + t]);
        cnt += ws_cnt[g * NLVL + l];
    }
    const float mean = (cnt > 0u) ? (s / (float)cnt) : 0.0f;
    const float mx   = (cnt > 0u) ? m : 0.0f;
    out[l * (2 * NDIM) + f]        = mean;
    out[l * (2 * NDIM) + NDIM + f] = mx;
}

extern "C" void kernel_launch(void* const* d_in, const int* in_sizes, int n_in,
                              void* d_out, int out_size, void* d_ws, size_t ws_size,
                              hipStream_t stream)
{
    const float* emb = (const float*)d_in[0];
    const int*   dep = (const int*)d_in[1];
    float*       out = (float*)d_out;

    const int N = in_sizes[1];                       // 500000 nodes
    const size_t blk_bytes = (size_t)(2 * NLVL * NDIM + NLVL) * 4;
    int G = (int)(ws_size / blk_bytes);              // deterministic given ws_size
    if (G < 1)    G = 1;
    if (G > 1024) G = 1024;
    const int chunk = (N + G - 1) / G;

    float*    ws_sum = (float*)d_ws;
    float*    ws_max = ws_sum + (size_t)G * NLVL * NDIM;
    unsigned* ws_cnt = (unsigned*)(ws_max + (size_t)G * NLVL * NDIM);

    lvlpool_phase1<<<G, 256, 0, stream>>>(emb, dep, ws_sum, ws_max, ws_cnt, N, chunk);
    lvlpool_phase2<<<(NLVL * NDIM) / 256, 256, 0, stream>>>(ws_sum, ws_max, ws_cnt, out, G);
}